// KFLoss_82188494176556
// MI455X (gfx1250) — compile-verified
//
#include <hip/hip_runtime.h>

#ifndef __has_builtin
#define __has_builtin(x) 0
#endif

// Use the CDNA5 async global->LDS path when the toolchain exposes it.
#if defined(__AMDGCN__) && __has_builtin(__builtin_amdgcn_global_load_async_to_lds_b128)
#define KF_ASYNC 1
#else
#define KF_ASYNC 0
#endif

#define KF_BLOCK 256

#if KF_ASYNC
typedef int kf_int4 __attribute__((ext_vector_type(4)));
#endif

__device__ __forceinline__ float kf_safe_sqrt(float x) {
  return x > 0.0f ? sqrtf(x) : 0.0f;
}

__device__ __forceinline__ float kf_smooth_l1(float d) {
  const float beta = 1.0f / 9.0f;
  // 0.5*d*d/beta == 4.5*d*d
  return d < beta ? 4.5f * d * d : d - 0.5f * beta;
}

__device__ __forceinline__ void kf_sigma(float w, float h, float r,
                                         float& a, float& b, float& d) {
  w = fminf(fmaxf(w, 1e-7f), 1e7f);
  h = fminf(fmaxf(h, 1e-7f), 1e7f);
  float c = cosf(r);
  float s = sinf(r);
  float hw2 = 0.25f * w * w;
  float hh2 = 0.25f * h * h;
  a = c * c * hw2 + s * s * hh2;
  b = c * s * (hw2 - hh2);
  d = s * s * hw2 + c * c * hh2;
}

__device__ __forceinline__ void kf_core(float px, float py, float tx, float ty,
                                        float pw, float ph, float pr,
                                        float tw, float th2, float tr,
                                        float* __restrict__ out_k,
                                        float* __restrict__ out_l, int i) {
  float xy = kf_smooth_l1(fabsf(px - tx)) + kf_smooth_l1(fabsf(py - ty));

  float ap, bp, dp, at, bt, dt;
  kf_sigma(pw, ph, pr, ap, bp, dp);
  kf_sigma(tw, th2, tr, at, bt, dt);

  float Vbp = 4.0f * kf_safe_sqrt(ap * dp - bp * bp);
  float Vbt = 4.0f * kf_safe_sqrt(at * dt - bt * bt);

  float sa = ap + at, sb = bp + bt, sd = dp + dt;
  float inv_det = 1.0f / (sa * sd - sb * sb);
  float ia = sd * inv_det, ib = -sb * inv_det, idd = sa * inv_det;

  float k11 = ap * ia + bp * ib;
  float k12 = ap * ib + bp * idd;
  float k21 = bp * ia + dp * ib;
  float k22 = bp * ib + dp * idd;

  float na = ap - (k11 * ap + k12 * bp);
  float nb = bp - (k11 * bp + k12 * dp);
  float nc = bp - (k21 * ap + k22 * bp);
  float nd = dp - (k21 * bp + k22 * dp);

  float Vb = 4.0f * kf_safe_sqrt(na * nd - nb * nc);
  float kf = Vb / (Vbp + Vbt - Vb + 1e-6f);

  // Reference: KFIoU.astype(fp16); kf_loss = (1 - KFIoU) in fp16, widened.
  _Float16 kh = (_Float16)kf;
  float kloss = (float)(_Float16)((_Float16)1.0f - kh);
  float loss = fmaxf(xy + kloss, 0.0f);

  __builtin_nontemporal_store((float)kh, out_k + i);
  __builtin_nontemporal_store(loss, out_l + i);
}

__global__ __launch_bounds__(KF_BLOCK) void kfiou_kernel(
    const float* __restrict__ pred, const float* __restrict__ targ,
    const float* __restrict__ pdec, const float* __restrict__ tdec,
    float* __restrict__ out_k, float* __restrict__ out_l, int n) {
#if KF_ASYNC
  // Stage one 256-row tile of each input (5120 B each) into LDS with async
  // b128 copies: 320 chunks per array: 1 per thread, 2 for tid < 64.
  __shared__ __align__(16) float smem[4 * KF_BLOCK * 5];
  char* sbytes = (char*)smem;
  const int tid = threadIdx.x;
  const long long tile_byte = (long long)blockIdx.x * (KF_BLOCK * 20);
  const long long limit = (long long)n * 20 - 16;  // last valid 16B chunk start

  auto stage = [&](const float* __restrict__ src, int lds_base) {
    long long g0 = tile_byte + (long long)tid * 16;
    long long ga = g0 <= limit ? g0 : limit;  // clamp tail (rows >= n unused)
    __builtin_amdgcn_global_load_async_to_lds_b128(
        (__attribute__((address_space(1))) kf_int4*)((char*)src + ga),
        (__attribute__((address_space(3))) kf_int4*)(sbytes + lds_base +
                                                     tid * 16),
        0, 0);
    if (tid < 64) {  // chunks 256..319 of this array
      long long g1 = g0 + 4096;
      long long gb = g1 <= limit ? g1 : limit;
      __builtin_amdgcn_global_load_async_to_lds_b128(
          (__attribute__((address_space(1))) kf_int4*)((char*)src + gb),
          (__attribute__((address_space(3))) kf_int4*)(sbytes + lds_base +
                                                       4096 + tid * 16),
          0, 0);
    }
  };
  stage(pred, 0);
  stage(targ, 5120);
  stage(pdec, 10240);
  stage(tdec, 15360);

#if __has_builtin(__builtin_amdgcn_s_wait_asynccnt)
  __builtin_amdgcn_s_wait_asynccnt(0);
#else
  asm volatile("s_wait_asynccnt 0" ::: "memory");
#endif
  __syncthreads();

  int i = blockIdx.x * KF_BLOCK + tid;
  if (i < n) {
    const float* sp = smem + tid * 5;           // pred rows
    const float* st = smem + 1280 + tid * 5;    // target rows
    const float* p5 = smem + 2560 + tid * 5;    // pred_decode rows
    const float* t5 = smem + 3840 + tid * 5;    // targets_decode rows
    kf_core(sp[0], sp[1], st[0], st[1],
            p5[2], p5[3], p5[4],
            t5[2], t5[3], t5[4],
            out_k, out_l, i);
  }
#else
  // Fallback: direct fully-coalesced streaming path with non-temporal hints.
  int i = blockIdx.x * blockDim.x + threadIdx.x;
  if (i < n) {
    long long b = (long long)i * 5;
    float px = __builtin_nontemporal_load(pred + b);
    float py = __builtin_nontemporal_load(pred + b + 1);
    float tx = __builtin_nontemporal_load(targ + b);
    float ty = __builtin_nontemporal_load(targ + b + 1);
    float pw = __builtin_nontemporal_load(pdec + b + 2);
    float ph = __builtin_nontemporal_load(pdec + b + 3);
    float pr = __builtin_nontemporal_load(pdec + b + 4);
    float tw = __builtin_nontemporal_load(tdec + b + 2);
    float t2 = __builtin_nontemporal_load(tdec + b + 3);
    float tr = __builtin_nontemporal_load(tdec + b + 4);
    kf_core(px, py, tx, ty, pw, ph, pr, tw, t2, tr, out_k, out_l, i);
  }
#endif
}

extern "C" void kernel_launch(void* const* d_in, const int* in_sizes, int n_in,
                              void* d_out, int out_size, void* d_ws,
                              size_t ws_size, hipStream_t stream) {
  (void)n_in;
  (void)out_size;
  (void)d_ws;
  (void)ws_size;
  const float* pred = (const float*)d_in[0];
  const float* targ = (const float*)d_in[1];
  const float* pdec = (const float*)d_in[2];
  const float* tdec = (const float*)d_in[3];
  int n = in_sizes[0] / 5;  // (N,5) float32
  float* out_k = (float*)d_out;      // KFIoU (fp16-rounded, stored as float)
  float* out_l = out_k + n;          // loss
  int blocks = (n + KF_BLOCK - 1) / KF_BLOCK;
  kfiou_kernel<<<blocks, KF_BLOCK, 0, stream>>>(pred, targ, pdec, tdec, out_k,
                                                out_l, n);
}